// LightGCN_60954175865427
// MI455X (gfx1250) — compile-verified
//
#include <hip/hip_runtime.h>

#define USER_NUM 100000
#define ITEM_NUM 50000
#define N_NODES  150000
#define NNZ      1200000
#define BATCH    4096
#define HIDDEN   64

typedef __attribute__((ext_vector_type(2))) float v2f;
typedef __attribute__((ext_vector_type(8))) float v8f;

__device__ __forceinline__ void fadd_agent(float* p, float v) {
    // relaxed + agent scope -> backend emits global_atomic_add_f32 (no CAS loop)
    __hip_atomic_fetch_add(p, v, __ATOMIC_RELAXED, __HIP_MEMORY_SCOPE_AGENT);
}

// cur = emb (concat), sum = 0.25*emb, nxt = 0
__global__ void lgcn_init(const float* __restrict__ uemb, const float* __restrict__ iemb,
                          float* __restrict__ cur, float* __restrict__ sum,
                          float* __restrict__ nxt) {
    size_t i = (size_t)blockIdx.x * blockDim.x + threadIdx.x;
    const size_t total = (size_t)N_NODES * HIDDEN;
    if (i >= total) return;
    const size_t usz = (size_t)USER_NUM * HIDDEN;
    float e = (i < usz) ? uemb[i] : iemb[i - usz];
    cur[i] = e;
    sum[i] = 0.25f * e;
    nxt[i] = 0.0f;
}

// y[dst[e]] += w[e] * x[src[e]]  (edge-parallel; 16 threads per edge, float4 each)
__global__ void lgcn_spmm(const int* __restrict__ src, const int* __restrict__ dst,
                          const float* __restrict__ w, const float* __restrict__ x,
                          float* __restrict__ y) {
    size_t t = (size_t)blockIdx.x * blockDim.x + threadIdx.x;
    size_t e = t >> 4;
    int s = (int)(t & 15);
    if (e >= (size_t)NNZ) return;
    float we = w[e];
    const float4* xr = (const float4*)(x + (size_t)src[e] * HIDDEN);
    float4 v = xr[s];                       // global_load_b128, 256B-aligned rows
    float* yr = y + (size_t)dst[e] * HIDDEN + (size_t)s * 4;
    fadd_agent(yr + 0, we * v.x);           // hw f32 atomics, resolve in 192MB L2
    fadd_agent(yr + 1, we * v.y);
    fadd_agent(yr + 2, we * v.z);
    fadd_agent(yr + 3, we * v.w);
}

// sum += 0.25*nxt; zero old cur so it can serve as next layer's accumulator
__global__ void lgcn_axpy_clear(const float* __restrict__ nxt, float* __restrict__ sum,
                                float* __restrict__ cur) {
    size_t i = (size_t)blockIdx.x * blockDim.x + threadIdx.x;
    const size_t total = (size_t)N_NODES * HIDDEN;
    if (i >= total) return;
    sum[i] += 0.25f * nxt[i];
    cur[i] = 0.0f;
}

// Last layer: no clear needed (saves 38.4MB of dead stores)
__global__ void lgcn_axpy(const float* __restrict__ nxt, float* __restrict__ sum) {
    size_t i = (size_t)blockIdx.x * blockDim.x + threadIdx.x;
    const size_t total = (size_t)N_NODES * HIDDEN;
    if (i >= total) return;
    sum[i] += 0.25f * nxt[i];
}

// One wave (32 lanes) per tile of 16 batch entries.
// C(16x16,f32) = A(16x64) * B(64x16) accumulated via 16x V_WMMA_F32_16X16X4_F32;
// gamma = diag(C); out = sigmoid(gamma).
__global__ void lgcn_gamma(const float* __restrict__ semb, const int* __restrict__ user,
                           const int* __restrict__ item, float* __restrict__ out) {
    __shared__ float diag[16];
    const int tile = blockIdx.x;          // BATCH/16 tiles
    const int lane = threadIdx.x;         // 0..31, wave32, EXEC all-ones through WMMA
    const int m    = lane & 15;           // A row / B col handled by this lane
    const int koff = (lane >> 4) << 1;    // lanes 0-15: K+{0,1}; lanes 16-31: K+{2,3}

    const float* U = semb + (size_t)user[tile * 16 + m] * HIDDEN;
    const float* V = semb + ((size_t)USER_NUM + (size_t)item[tile * 16 + m]) * HIDDEN;

    v8f c = {};
#pragma unroll
    for (int kc = 0; kc < HIDDEN / 4; ++kc) {
        int k = kc * 4 + koff;
        v2f a; a.x = U[k]; a.y = U[k + 1];   // A 16x4 layout: VGPR0=K0/K2, VGPR1=K1/K3
        v2f b; b.x = V[k]; b.y = V[k + 1];   // B 4x16 layout mirrors A
        c = __builtin_amdgcn_wmma_f32_16x16x4_f32(
                /*neg_a=*/false, a, /*neg_b=*/false, b,
                /*c_mod=*/(short)0, c, /*reuse_a=*/false, /*reuse_b=*/false);
    }

    // Diagonal element (d,d): VGPR d&7, lane d (d<8) or lane d+16 (d>=8).
    bool has = (lane < 8) || (lane >= 24);
    int  r   = (lane < 8) ? lane : (lane - 24);   // accumulator VGPR index
    int  d   = (lane < 8) ? lane : (lane - 16);   // diagonal index
    float dv = 0.0f;
#pragma unroll
    for (int i2 = 0; i2 < 8; ++i2)
        if (r == i2) dv = c[i2];
    if (has) diag[d] = dv;
    __syncthreads();

    if (lane < 16) {
        float g = diag[lane];
        out[tile * 16 + lane] = 1.0f / (1.0f + __expf(-g));
    }
}

extern "C" void kernel_launch(void* const* d_in, const int* in_sizes, int n_in,
                              void* d_out, int out_size, void* d_ws, size_t ws_size,
                              hipStream_t stream) {
    const int*   user = (const int*)d_in[0];
    const int*   item = (const int*)d_in[1];
    const int*   src  = (const int*)d_in[2];
    const int*   dst  = (const int*)d_in[3];
    const float* ew   = (const float*)d_in[4];
    const float* ue   = (const float*)d_in[5];
    const float* ie   = (const float*)d_in[6];
    float* out = (float*)d_out;

    const size_t nf = (size_t)N_NODES * HIDDEN;   // 9.6M floats per buffer
    float* cur = (float*)d_ws;
    float* nxt = cur + nf;
    float* sum = nxt + nf;

    const int TB = 256;
    const unsigned gElem = (unsigned)((nf + TB - 1) / TB);
    lgcn_init<<<gElem, TB, 0, stream>>>(ue, ie, cur, sum, nxt);

    const unsigned gSp = (unsigned)(((size_t)NNZ * 16 + TB - 1) / TB);
    // layers 1..2: accumulate and recycle cur as next accumulator
    for (int layer = 1; layer < 3; ++layer) {
        lgcn_spmm<<<gSp, TB, 0, stream>>>(src, dst, ew, cur, nxt);
        lgcn_axpy_clear<<<gElem, TB, 0, stream>>>(nxt, sum, cur);
        float* t = cur; cur = nxt; nxt = t;       // host-side swap, capture-safe
    }
    // layer 3: no recycle needed
    lgcn_spmm<<<gSp, TB, 0, stream>>>(src, dst, ew, cur, nxt);
    lgcn_axpy<<<gElem, TB, 0, stream>>>(nxt, sum);

    lgcn_gamma<<<BATCH / 16, 32, 0, stream>>>(sum, user, item, out);
}